// NonMaximaSuppression3d_35021163332235
// MI455X (gfx1250) — compile-verified
//
#include <hip/hip_runtime.h>
#include <cstdint>

// Problem geometry (fixed by the reference):  x : (4,4,32,256,256) f32
#define DW 256              // W
#define DH 256              // H
#define DD 32               // D
#define TX 32               // output tile width  (== lanes per row)
#define TY 16               // output tile height
#define TZ 8                // output tile depth
#define LX (TX + 2)         // 34  LDS tile (with halo)
#define LY (TY + 2)         // 18
#define LZ (TZ + 2)         // 10

typedef uint32_t u32x4 __attribute__((ext_vector_type(4)));
typedef int      i32x4 __attribute__((ext_vector_type(4)));
typedef int      i32x8 __attribute__((ext_vector_type(8)));

__global__ __launch_bounds__(256)
void nms3d_tdm_kernel(const float* __restrict__ xin, float* __restrict__ out)
{
    __shared__ float tile[LZ * LY * LX];     // 24,480 B — sole LDS object → offset 0

    const int tileX = blockIdx.x;            // 0..7
    const int tileY = blockIdx.y;            // 0..15
    const int slab  = blockIdx.z & 3;        // 0..3   (32 / TZ)
    const int vol   = blockIdx.z >> 2;       // 0..15  (B*CH)

    const int x0 = tileX * TX;
    const int y0 = tileY * TY;
    const int z0 = slab  * TZ;

    // Clamp tile start on the low side (boundary outputs are 0 regardless);
    // high side is handled by TDM out-of-tensor zero fill.
    const int sx = (x0 > 0) ? x0 - 1 : 0;
    const int sy = (y0 > 0) ? y0 - 1 : 0;
    const int sz = (z0 > 0) ? z0 - 1 : 0;
    const int offx = x0 - sx;                // 0 or 1
    const int offy = y0 - sy;
    const int offz = z0 - sz;

    // ---- TDM: DMA the (LX x LY x LZ) halo tile into LDS (wave 0 issues) ----
    if ((threadIdx.x >> 5) == 0) {
        const uint64_t gaddr = (uint64_t)(uintptr_t)xin
            + 4ull * ((uint64_t)vol * (DD * DH * DW)
                    + (uint64_t)sz  * (DH * DW)
                    + (uint64_t)sy  *  DW
                    + (uint64_t)sx);

        // D# group 0: count=1 | lds_addr=0 | global_addr | type=2
        u32x4 g0;
        g0[0] = 1u;
        g0[1] = 0u;
        g0[2] = (uint32_t)(gaddr & 0xFFFFFFFFu);
        g0[3] = (uint32_t)((gaddr >> 32) & 0x01FFFFFFu) | (2u << 30);

        // D# group 1: data_size=4B; tensor dims/strides; tile dims
        const uint32_t d0  = DW;                 // tensor_dim0
        const uint32_t d1  = DH;                 // tensor_dim1
        const uint32_t s0  = DW;                 // tensor_dim0_stride
        const uint32_t s1  = DW * DH;            // tensor_dim1_stride (65536)
        i32x8 g1;
        g1[0] = (int)(2u << 16);                               // data_size = 2 (4 bytes)
        g1[1] = (int)((d0 & 0xFFFFu) << 16);                   // dim0[15:0]  -> bits 63:48
        g1[2] = (int)((d0 >> 16) | ((d1 & 0xFFFFu) << 16));    // dim0 hi | dim1 lo
        g1[3] = (int)((d1 >> 16) | ((uint32_t)LX << 16));      // dim1 hi | tile_dim0
        g1[4] = (int)((uint32_t)LY | ((uint32_t)LZ << 16));    // tile_dim1 | tile_dim2
        g1[5] = (int)s0;                                       // dim0_stride[31:0]
        g1[6] = (int)((s1 & 0xFFFFu) << 16);                   // d0s[47:32]=0 | d1s[15:0]
        g1[7] = (int)(s1 >> 16);                               // d1s[47:16]

        // D# group 2: tensor_dim2=D; tile_dim3=0 (3D tile)
        i32x4 g2;
        g2[0] = DD;
        g2[1] = 1;                        // tensor_dim3 (unused)
        g2[2] = (int)(DW * DH * DD);      // tensor_dim2_stride lo (unused, tile_dim3=0)
        g2[3] = 0;                        // d2s hi | tile_dim3 = 0

        i32x4 g3 = {0, 0, 0, 0};          // group 3 unused (no dim4)

#if __clang_major__ >= 23
        i32x8 gpad = {0, 0, 0, 0, 0, 0, 0, 0};
        __builtin_amdgcn_tensor_load_to_lds(g0, g1, g2, g3, gpad, 0);
#else
        __builtin_amdgcn_tensor_load_to_lds(g0, g1, g2, g3, 0);
#endif
        __builtin_amdgcn_s_wait_tensorcnt(0);
    }
    __syncthreads();

    // The TDM intrinsic carries the LDS destination only as descriptor *data*,
    // so LLVM cannot see that tile[] was written. Force the optimizer to treat
    // tile[] as modified, otherwise every ds_load below folds to undef.
    {
        float* tp = &tile[0];
        asm volatile("" : "+r"(tp) : : "memory");
    }

    // ---- Compute: rolling 3-plane register window along z ----
    const int tid = threadIdx.x;
    const int lx  = tid & (TX - 1);       // lane = x  -> coalesced stores
    const int lyb = tid >> 5;             // 0..7

    const int gx = x0 + lx;
    const bool bx = (gx == 0) | (gx == DW - 1);

    int xm = lx + offx - 1; if (xm < 0) xm = 0;
    const int xc = lx + offx;
    const int xp = lx + offx + 1;

    float* const ovol = out + (uint64_t)vol * (DD * DH * DW);

    for (int yi = 0; yi < TY / 8; ++yi) {
        const int ly = lyb + yi * 8;
        const int gy = y0 + ly;
        const bool by = (gy == 0) | (gy == DH - 1);

        int ym = ly + offy - 1; if (ym < 0) ym = 0;
        const int yc = ly + offy;
        const int yp = ly + offy + 1;

        float m[9], c[9], p[9];
        auto plane = [&](float* dst, int zz) {
            const float* b = &tile[zz * (LY * LX)];
            dst[0] = b[ym * LX + xm]; dst[1] = b[ym * LX + xc]; dst[2] = b[ym * LX + xp];
            dst[3] = b[yc * LX + xm]; dst[4] = b[yc * LX + xc]; dst[5] = b[yc * LX + xp];
            dst[6] = b[yp * LX + xm]; dst[7] = b[yp * LX + xc]; dst[8] = b[yp * LX + xp];
        };

        int zinit = offz - 1; if (zinit < 0) zinit = 0;
        plane(m, zinit);
        plane(c, offz);

        #pragma unroll
        for (int lz = 0; lz < TZ; ++lz) {
            plane(p, lz + offz + 1);

            const int gz = z0 + lz;
            const bool bz = (gz == 0) | (gz == DD - 1);
            const float ctr = c[4];

            // max over the 26 neighbors (strict > test afterwards)
            float nmax = m[0];
            #pragma unroll
            for (int k = 1; k < 9; ++k) nmax = fmaxf(nmax, m[k]);
            #pragma unroll
            for (int k = 0; k < 9; ++k) if (k != 4) nmax = fmaxf(nmax, c[k]);
            #pragma unroll
            for (int k = 0; k < 9; ++k) nmax = fmaxf(nmax, p[k]);

            const bool keep = (!(bx | by | bz)) && (ctr > nmax);
            ovol[((uint64_t)gz * DH + gy) * DW + gx] = keep ? ctr : 0.0f;

            #pragma unroll
            for (int k = 0; k < 9; ++k) { m[k] = c[k]; c[k] = p[k]; }
        }
    }
}

extern "C" void kernel_launch(void* const* d_in, const int* in_sizes, int n_in,
                              void* d_out, int out_size, void* d_ws, size_t ws_size,
                              hipStream_t stream)
{
    (void)n_in; (void)out_size; (void)d_ws; (void)ws_size;
    const float* xin = (const float*)d_in[0];
    float*       out = (float*)d_out;

    const int nvol = in_sizes[0] / (DD * DH * DW);   // B*CH = 16
    dim3 grid(DW / TX, DH / TY, (DD / TZ) * nvol);   // (8, 16, 64)
    dim3 block(256, 1, 1);
    nms3d_tdm_kernel<<<grid, block, 0, stream>>>(xin, out);
}